// HungarianMatcher_70128226009302
// MI455X (gfx1250) — compile-verified
//
#include <hip/hip_runtime.h>

typedef __attribute__((ext_vector_type(2))) float v2f;
typedef __attribute__((ext_vector_type(8))) float v8f;

#define KP 96          // class dim (91) padded to multiple of 16
#define TILE 16
#define WPB 4          // waves per block (block covers 16 rows x 64 cols)

__device__ __forceinline__ unsigned fkey(float f) {
  unsigned b = __float_as_uint(f);
  return (b & 0x80000000u) ? ~b : (b | 0x80000000u);   // order-preserving map
}
__device__ __forceinline__ float funkey(unsigned k) {
  return (k & 0x80000000u) ? __uint_as_float(k & 0x7fffffffu) : __uint_as_float(~k);
}
__device__ __forceinline__ bool finitef(float v) {
  return ((__float_as_uint(v) >> 23) & 0xffu) != 0xffu;
}

// Async global->LDS copy of 16 bytes (gfx1250 GLOBAL_LOAD_ASYNC_TO_LDS_B128).
// ldsAddr: LDS byte address (low 32 bits of generic pointer == LDS offset).
__device__ __forceinline__ void async_copy_b128(unsigned ldsAddr,
                                                unsigned long long gAddr) {
  asm volatile("global_load_async_to_lds_b128 %0, %1, off"
               :: "v"(ldsAddr), "v"(gAddr) : "memory");
}
__device__ __forceinline__ void wait_async0() {
  asm volatile("s_wait_asynccnt 0x0" ::: "memory");
}

// ---------------- Kernel 1: focal class-cost table [rows, KP] ----------------
__global__ void k_class_table(const float* __restrict__ logits,
                              float* __restrict__ table, int rows, int nc) {
  int idx = blockIdx.x * blockDim.x + threadIdx.x;
  if (idx >= rows * KP) return;
  int q = idx / KP, c = idx - q * KP;
  float v = 0.f;
  if (c < nc) {
    float x  = logits[q * nc + c];
    float p  = 1.f / (1.f + __expf(-x));
    float ax = fabsf(x);
    float l1p = log1pf(__expf(-ax));
    float sp_pos = fmaxf(x, 0.f) + l1p;    // softplus(x)  = -log_sigmoid(-x)
    float sp_neg = fmaxf(-x, 0.f) + l1p;   // softplus(-x) = -log_sigmoid(x)
    float pos = 0.25f * (1.f - p) * (1.f - p) * sp_neg;
    float neg = 0.75f * p * p * sp_pos;
    v = pos - neg;
  }
  table[idx] = v;
}

// ----- Kernel 2: one-hot B [KP, nt], target xyxy+area, init max-accum -------
__global__ void k_targets(const int* __restrict__ ids, const float* __restrict__ tb,
                          float* __restrict__ onehot, float* __restrict__ tpre,
                          unsigned* __restrict__ umax, int nt, int nc) {
  int idx = blockIdx.x * blockDim.x + threadIdx.x;
  if (idx == 0) *umax = fkey(-3.402823466e38f);
  if (idx < KP * nt) {
    int c = idx / nt, t = idx - c * nt;
    onehot[idx] = (c < nc && ids[t] == c) ? 1.f : 0.f;
  }
  if (idx < nt) {
    float cx = tb[idx * 4 + 0], cy = tb[idx * 4 + 1];
    float w  = tb[idx * 4 + 2], h  = tb[idx * 4 + 3];
    float x0 = cx - 0.5f * w, y0 = cy - 0.5f * h;
    float x1 = cx + 0.5f * w, y1 = cy + 0.5f * h;
    tpre[idx * 8 + 0] = x0; tpre[idx * 8 + 1] = y0;
    tpre[idx * 8 + 2] = x1; tpre[idx * 8 + 3] = y1;
    tpre[idx * 8 + 4] = (x1 - x0) * (y1 - y0);
    tpre[idx * 8 + 5] = 0.f; tpre[idx * 8 + 6] = 0.f; tpre[idx * 8 + 7] = 0.f;
  }
}

// --------- Kernel 3: fused cost tile (WMMA gather-GEMM + L1 + GIoU) ---------
__global__ void __launch_bounds__(32 * WPB)
k_cost(const float* __restrict__ table, const float* __restrict__ onehot,
       const float* __restrict__ pboxes, const float* __restrict__ tbb,
       const float* __restrict__ tpre, float* __restrict__ out,
       unsigned* __restrict__ umax, int nt, int colTiles, int colGroups) {
  __shared__ float sTab[TILE * KP];       // 16 x 96 A tile (class-cost rows)
  __shared__ float sB[KP * 64];           // 96 x 64 one-hot tile (block cols)
  __shared__ float sBox[TILE * 4];        // 16 pred boxes (cxcywh)

  int rowTile = blockIdx.x / colGroups;
  int group   = blockIdx.x - rowTile * colGroups;
  int wave    = threadIdx.x >> 5;
  int lane    = threadIdx.x & 31;
  int rbase   = rowTile * TILE;
  int cbase   = group * (TILE * WPB);

  __builtin_prefetch(&tpre[(size_t)cbase * 8], 0, 1);

  // --- async-stage A tile: 16 contiguous rows of KP floats -> 384 x b128 ---
  const float* tsrc = table + (size_t)rbase * KP;
  for (int j = threadIdx.x; j < (TILE * KP) / 4; j += blockDim.x) {
    async_copy_b128((unsigned)(uintptr_t)&sTab[j * 4],
                    (unsigned long long)(uintptr_t)(tsrc + j * 4));
  }
  // --- async-stage B tile: 96 rows x 64 cols (16 x b128 per row) ---
  for (int j = threadIdx.x; j < KP * 16; j += blockDim.x) {
    int krow = j >> 4;
    int c4   = (j & 15) * 4;
    if (cbase + c4 + 3 < nt) {
      async_copy_b128((unsigned)(uintptr_t)&sB[krow * 64 + c4],
                      (unsigned long long)(uintptr_t)
                          (onehot + (size_t)krow * nt + cbase + c4));
    }
  }
  // --- pred boxes (tiny, plain path) ---
  for (int i = threadIdx.x; i < TILE * 4; i += blockDim.x)
    sBox[i] = pboxes[(size_t)rbase * 4 + i];

  wait_async0();
  __syncthreads();

  int colTile = group * WPB + wave;
  if (colTile >= colTiles) return;          // wave-uniform

  int n     = lane & 15;                    // N column / A row of this lane
  int khalf = (lane >> 4) * 2;              // K sub-offset per half-wave
  int cb    = wave * TILE + n;              // column within block tile
  int t     = cbase + cb;                   // global target column

  v8f acc = {};
#pragma unroll
  for (int k = 0; k < KP; k += 4) {
    int k0 = k + khalf;
    v2f a = *(const v2f*)&sTab[n * KP + k0];          // ds_load_b64, imm offset
    v2f b;
    b.x = sB[k0 * 64 + cb];                           // ds_load_b32, imm offset
    b.y = sB[(k0 + 1) * 64 + cb];
    acc = __builtin_amdgcn_wmma_f32_16x16x4_f32(
        false, a, false, b, (short)0, acc, false, false);
  }

  // per-lane target data (fixed N per lane)
  float tx0 = tpre[t * 8 + 0], ty0 = tpre[t * 8 + 1];
  float tx1 = tpre[t * 8 + 2], ty1 = tpre[t * 8 + 3];
  float tarea = tpre[t * 8 + 4];
  float tcx = tbb[t * 4 + 0], tcy = tbb[t * 4 + 1];
  float tw  = tbb[t * 4 + 2], th  = tbb[t * 4 + 3];

  float localmax = -3.402823466e38f;
  int mbase = (lane >> 4) * 8;              // C frag: VGPR r -> M = r or r+8
#pragma unroll
  for (int r = 0; r < 8; ++r) {
    int m = mbase + r;
    float pcx = sBox[m * 4 + 0], pcy = sBox[m * 4 + 1];
    float pw  = sBox[m * 4 + 2], ph  = sBox[m * 4 + 3];
    float p0 = pcx - 0.5f * pw, p1 = pcy - 0.5f * ph;
    float p2 = pcx + 0.5f * pw, p3 = pcy + 0.5f * ph;
    float parea = (p2 - p0) * (p3 - p1);

    float iw = fmaxf(fminf(p2, tx1) - fmaxf(p0, tx0), 0.f);
    float ih = fmaxf(fminf(p3, ty1) - fmaxf(p1, ty0), 0.f);
    float inter = iw * ih;
    float uni = parea + tarea - inter;
    float iou = inter / uni;
    float ew = fmaxf(fmaxf(p2, tx1) - fminf(p0, tx0), 0.f);
    float eh = fmaxf(fmaxf(p3, ty1) - fminf(p1, ty0), 0.f);
    float ae = ew * eh;
    float giou = iou - (ae - uni) / ae;

    float l1 = fabsf(pcx - tcx) + fabsf(pcy - tcy) +
               fabsf(pw - tw) + fabsf(ph - th);
    float c = l1 + acc[r] - giou;
    out[(size_t)(rbase + m) * nt + t] = c;
    if (finitef(c)) localmax = fmaxf(localmax, c);
  }

  for (int off = 16; off > 0; off >>= 1)
    localmax = fmaxf(localmax, __shfl_xor(localmax, off, 32));
  if (lane == 0) atomicMax(umax, fkey(localmax));
}

// ----------- Kernel 4: sanitize non-finite with 2*max (or 1e6) --------------
__global__ void k_sanitize(float* __restrict__ out,
                           const unsigned* __restrict__ umax, size_t nelem) {
  size_t i = (size_t)blockIdx.x * blockDim.x + threadIdx.x;
  if (i >= nelem) return;
  float v = out[i];
  if (!finitef(v)) {
    float mx = funkey(*umax);
    out[i] = (mx > 0.f) ? mx * 2.f : 1e6f;
  }
}

extern "C" void kernel_launch(void* const* d_in, const int* in_sizes, int n_in,
                              void* d_out, int out_size, void* d_ws, size_t ws_size,
                              hipStream_t stream) {
  const float* pred_logits = (const float*)d_in[0];   // [bs,nq,nc] f32
  const float* pred_boxes  = (const float*)d_in[1];   // [bs,nq,4]  f32
  const int*   tgt_ids     = (const int*)d_in[2];     // [nt]       int
  const float* tgt_bbox    = (const float*)d_in[3];   // [nt,4]     f32
  float* out = (float*)d_out;

  int rows = in_sizes[1] / 4;        // bs*nq = 14400
  int nc   = in_sizes[0] / rows;     // 91
  int nt   = in_sizes[2];            // 1600

  char* ws = (char*)d_ws;
  float* table  = (float*)ws;                    size_t off = (size_t)rows * KP * 4;
  float* onehot = (float*)(ws + off);            off += (size_t)KP * nt * 4;
  float* tpre   = (float*)(ws + off);            off += (size_t)nt * 8 * 4;
  unsigned* umax = (unsigned*)(ws + off);

  {
    int total = rows * KP;
    k_class_table<<<(total + 255) / 256, 256, 0, stream>>>(pred_logits, table, rows, nc);
  }
  {
    int total = KP * nt;
    k_targets<<<(total + 255) / 256, 256, 0, stream>>>(tgt_ids, tgt_bbox, onehot,
                                                       tpre, umax, nt, nc);
  }
  {
    int rowTiles  = rows / TILE;                       // 900
    int colTiles  = nt / TILE;                         // 100
    int colGroups = (colTiles + WPB - 1) / WPB;        // 25
    k_cost<<<rowTiles * colGroups, 32 * WPB, 0, stream>>>(
        table, onehot, pred_boxes, tgt_bbox, tpre, out, umax,
        nt, colTiles, colGroups);
  }
  {
    size_t nelem = (size_t)rows * nt;
    k_sanitize<<<(unsigned)((nelem + 255) / 256), 256, 0, stream>>>(out, umax, nelem);
  }
}